// PseudoSpectra2d_5291399708937
// MI455X (gfx1250) — compile-verified
//
#include <hip/hip_runtime.h>
#include <hip/hip_bf16.h>

// PseudoSpectra2d on gfx1250 (MI455X), wave32.
// f32 WMMA 16x16x4 for all GEMMs; TDM (tensor_load_to_lds, TENSORcnt) stages
// u-tiles in k1; global_load_async_to_lds_b128 (ASYNCcnt) stages out-tiles in k3.
// B=16, CIN=32, COUT=32, H=W=128, M1=M2=64, KH=KW=3, TRIL=1, P1=P2=64.

typedef __attribute__((ext_vector_type(2))) float v2f;
typedef __attribute__((ext_vector_type(8))) float v8f;
typedef __attribute__((ext_vector_type(4))) unsigned int u32x4;
typedef __attribute__((ext_vector_type(8))) unsigned int u32x8;

#define BATCH 16
#define CIN   32
#define COUT  32
#define HW    128
#define MM    64
#define KDIM  (CIN * 3 * 3)       // 288
#define X_N   (MM * MM)           // 4096

__device__ __forceinline__ v8f wmma4(v2f a, v2f b, v8f c) {
    // D = A(16x4,f32) * B(4x16,f32) + C(16x16,f32)
    return __builtin_amdgcn_wmma_f32_16x16x4_f32(
        /*neg_a=*/false, a, /*neg_b=*/false, b,
        /*c_mod=*/(short)0, c, /*reuse_a=*/false, /*reuse_b=*/false);
}

// A fragment: row-major A[lda], tile origin (row0, kbase).
// lane l holds A[row0 + (l&15)][kbase + 2*(l>>4) + {0,1}]
__device__ __forceinline__ v2f load_a_rm(const float* __restrict__ A, int lda,
                                         int row0, int kbase, int lane) {
    int m  = lane & 15;
    int ks = kbase + ((lane >> 4) << 1);
    const float* p = A + (size_t)(row0 + m) * lda + ks;
    v2f r; r.x = p[0]; r.y = p[1];
    return r;
}

// B fragment: row-major B[K][N], leading dim ldb, tile origin (kbase, col0).
// lane l holds B[kbase + 2*(l>>4) + {0,1}][col0 + (l&15)]
__device__ __forceinline__ v2f load_b_rm(const float* __restrict__ Bm, int ldb,
                                         int kbase, int col0, int lane) {
    int n  = lane & 15;
    int ks = kbase + ((lane >> 4) << 1);
    v2f r;
    r.x = Bm[(size_t)ks       * ldb + col0 + n];
    r.y = Bm[(size_t)(ks + 1) * ldb + col0 + n];
    return r;
}

// C/D 16x16: lane l -> column (l&15); VGPR r -> row r + 8*(l>>4)
__device__ __forceinline__ void store_c_rm(float* __restrict__ C, int ldc,
                                           int row0, int col0, int lane, v8f acc) {
    int n  = lane & 15;
    int mo = (lane >> 4) << 3;
#pragma unroll
    for (int r = 0; r < 8; ++r)
        C[(size_t)(row0 + mo + r) * ldc + col0 + n] = acc[r];
}

// ---------------------------------------------------------------------------
// TDM: 1-D tile load (nelems f32, nelems <= 65535) global -> LDS.
// D# built per CDNA5 ISA §8: group0 = {count|flags, lds_addr, gaddr_lo,
// gaddr_hi|type=2}; group1 = dims/strides (data_size=4B, tile==tensor, 1-D).
// Issue from ONE wave; follow with s_wait_tensorcnt + workgroup barrier.
// ---------------------------------------------------------------------------
__device__ __forceinline__ void tdm_load_f32_1d(float* lds_dst, const float* gsrc,
                                                unsigned nelems) {
    unsigned long long ga = (unsigned long long)(uintptr_t)gsrc;
    unsigned lds_off = (unsigned)(uintptr_t)lds_dst;   // low 32 bits = LDS offset

    u32x4 g0;
    g0[0] = 1u;                                        // count=1, user D#
    g0[1] = lds_off;                                   // lds_addr (bytes)
    g0[2] = (unsigned)(ga & 0xFFFFFFFFu);              // global_addr[31:0]
    g0[3] = (unsigned)((ga >> 32) & 0x1FFFFFFu)        // global_addr[56:32]
          | (2u << 30);                                // type=2 ("image")

    u32x8 g1;
    g1[0] = 2u << 16;                                  // data_size=2 (4 bytes)
    g1[1] = (nelems & 0xFFFFu) << 16;                  // tensor_dim0[15:0]
    g1[2] = (nelems >> 16) | (1u << 16);               // tensor_dim0[31:16], tensor_dim1=1
    g1[3] = (nelems & 0xFFFFu) << 16;                  // tile_dim0 = nelems
    g1[4] = 0u;                                        // tile_dim1=0 (1-D), tile_dim2=0
    g1[5] = nelems;                                    // tensor_dim0_stride[31:0]
    g1[6] = 0u;                                        // stride hi / dim1_stride lo
    g1[7] = 0u;
    asm volatile("tensor_load_to_lds %0, %1" :: "s"(g0), "s"(g1) : "memory");
}

// ---------------------------------------------------------------------------
// Kernel 1: bp[b,c,m,k] = sum_{h,w} u[b,c,h,w] * A1[m,h] * A2[k,w], m,k < 64
// One block (256 thr = 8 waves) per (b,c).
// TDM stages u[bc] (64 KB) into LDS; t = u * A2[:64]^T (LDS), bp = A1[:64] * t.
// Dynamic LDS: ulds[128*128] + tlds[128*64] = 96 KB.
// ---------------------------------------------------------------------------
__global__ void k1_spectral_fwd(const float* __restrict__ u,
                                const float* __restrict__ A1,
                                const float* __restrict__ A2,
                                float* __restrict__ bp) {
    extern __shared__ float smem[];
    float* ulds = smem;                 // [128][128]
    float* tlds = smem + HW * HW;       // [128][64]

    const int bc   = blockIdx.x;
    const int lane = threadIdx.x & 31;
    const int wave = threadIdx.x >> 5;
    const float* usrc = u + (size_t)bc * HW * HW;

    if (threadIdx.x < 32) {             // wave 0 issues the tensor DMA
        tdm_load_f32_1d(ulds, usrc, HW * HW);
        __builtin_amdgcn_s_wait_tensorcnt(0);
    }
    __syncthreads();

    // GEMM1: t[h][kd] = sum_w u[h][w] * A2[kd][w].  Tiles: 8(h) x 4(kd) = 32.
    for (int t = wave; t < 32; t += 8) {
        const int hr = (t >> 2) << 4;
        const int kc = (t & 3) << 4;
        v8f acc = {};
        const int n  = lane & 15;
        const int ko = (lane >> 4) << 1;
        for (int kb = 0; kb < HW; kb += 4) {
            v2f a = load_a_rm(ulds, HW, hr, kb, lane);       // ds_load
            // B[w][n] = A2[kc+n][w] (contiguous float2 along w of A2 row kc+n)
            const float* p = A2 + (size_t)(kc + n) * HW + kb + ko;
            v2f b; b.x = p[0]; b.y = p[1];
            acc = wmma4(a, b, acc);
        }
        store_c_rm(tlds, MM, hr, kc, lane, acc);
    }
    __syncthreads();

    // GEMM2: bp[m][kd] = sum_h A1[m][h] * t[h][kd].  Tiles: 4 x 4 = 16.
    float* bdst = bp + (size_t)bc * MM * MM;
    for (int t = wave; t < 16; t += 8) {
        const int mr = (t >> 2) << 4;
        const int kc = (t & 3) << 4;
        v8f acc = {};
        for (int kb = 0; kb < HW; kb += 4) {
            v2f a = load_a_rm(A1, HW, mr, kb, lane);
            v2f b = load_b_rm(tlds, MM, kb, kc, lane);       // ds_load
            acc = wmma4(a, b, acc);
        }
        store_c_rm(bdst, MM, mr, kc, lane, acc);
    }
}

// ---------------------------------------------------------------------------
// Kernel 2: out[b,o,x] = sum_i xpad[b,i,x] * W[i,o,x]   (x = m*64+k, i = 288)
// One wave per x: M=batch(16), N=cout(32 -> two 16-tiles), K=288.
// The 3x3 zero-padded unfold is folded into the A-fragment gather.
// Weights (151 MB) are streamed exactly once; consecutive-x blocks give
// full cacheline reuse through L2.
// ---------------------------------------------------------------------------
__global__ void k2_pointwise(const float* __restrict__ bp,
                             const float* __restrict__ W,
                             float* __restrict__ outm) {
    const int lane = threadIdx.x & 31;
    const int wave = threadIdx.x >> 5;
    const int x    = (blockIdx.x << 3) + wave;      // 0..4095
    const int mrow = x >> 6;
    const int kcol = x & 63;
    const int ln   = lane & 15;                     // batch row (A) / cout col (B)
    const int ko   = (lane >> 4) << 1;              // lane's K sub-offset

    v8f acc0 = {}, acc1 = {};
    for (int kk = 0; kk < KDIM; kk += 4) {
        // A fragment: xpad[b=ln, i=kk+ko+{0,1}, x] with zero-padded halo
        v2f a;
#pragma unroll
        for (int c2 = 0; c2 < 2; ++c2) {
            const int i   = kk + ko + c2;
            const int cch = i / 9;
            const int r   = i - cch * 9;
            const int di  = r / 3;
            const int dj  = r - di * 3;
            const int mi  = mrow + di - 1;
            const int ki  = kcol + dj - 1;
            float v = 0.0f;
            if (mi >= 0 && mi < MM && ki >= 0 && ki < MM)
                v = bp[(((size_t)ln * CIN + cch) * MM + mi) * MM + ki];
            if (c2 == 0) a.x = v; else a.y = v;
        }
        // B fragments: W[(i)*32 + o][x], o = ln and 16+ln
        const size_t base0 = ((size_t)(kk + ko) * COUT) * X_N + x;
        v2f b0, b1;
        b0.x = W[base0 + (size_t)ln * X_N];
        b0.y = W[base0 + (size_t)(COUT + ln) * X_N];
        b1.x = W[base0 + (size_t)(16 + ln) * X_N];
        b1.y = W[base0 + (size_t)(COUT + 16 + ln) * X_N];
        acc0 = wmma4(a, b0, acc0);
        acc1 = wmma4(a, b1, acc1);
    }
    const int mo = (lane >> 4) << 3;
#pragma unroll
    for (int r = 0; r < 8; ++r) {
        const int b = mo + r;
        outm[((size_t)b * COUT + ln)      * X_N + x] = acc0[r];
        outm[((size_t)b * COUT + 16 + ln) * X_N + x] = acc1[r];
    }
}

// ---------------------------------------------------------------------------
// Kernel 3: y[b,c,h,w] = sum_{m,k} out[b,c,m,k] * B1[h,m] * B2[w,k]
// One block per (b,c). out[bc] (16 KB) staged into LDS with per-lane
// global_load_async_to_lds_b128 (ASYNCcnt). tmp = out * B2^T, y = B1 * tmp.
// Dynamic LDS: olds[64*64] + tmp[64*128] = 48 KB.
// ---------------------------------------------------------------------------
__global__ void k3_spectral_bwd(const float* __restrict__ outm,
                                const float* __restrict__ B1,
                                const float* __restrict__ B2,
                                float* __restrict__ y) {
    extern __shared__ float smem[];
    float* olds = smem;                 // [64][64]
    float* tmpl = smem + MM * MM;       // [64][128]

    const int bc   = blockIdx.x;
    const int lane = threadIdx.x & 31;
    const int wave = threadIdx.x >> 5;
    const float* osrc = outm + (size_t)bc * MM * MM;

    // Async-stage out[bc] into LDS: 4096 f32 = 1024 x b128, 4 per thread.
    {
        const unsigned obase = (unsigned)(uintptr_t)olds;
        for (int r = threadIdx.x; r < (MM * MM / 4); r += 256) {
            const unsigned loff = obase + (unsigned)r * 16u;
            const float* gp = osrc + (size_t)r * 4;
            asm volatile("global_load_async_to_lds_b128 %0, %1, off"
                         :: "v"(loff), "v"(gp) : "memory");
        }
        asm volatile("s_wait_asynccnt 0x0" ::: "memory");
    }
    __syncthreads();

    // GEMM1: tmp[m][w] = sum_k out[m][k] * B2[w][k].  Tiles: 4(m) x 8(w) = 32.
    for (int t = wave; t < 32; t += 8) {
        const int mr = (t >> 3) << 4;
        const int wc = (t & 7) << 4;
        v8f acc = {};
        const int n  = lane & 15;
        const int ko = (lane >> 4) << 1;
        for (int kb = 0; kb < MM; kb += 4) {
            v2f a = load_a_rm(olds, MM, mr, kb, lane);       // ds_load
            // B[k][n] = B2[wc+n][k] (contiguous float2 along k of B2 row wc+n)
            const float* p = B2 + (size_t)(wc + n) * MM + kb + ko;
            v2f b; b.x = p[0]; b.y = p[1];
            acc = wmma4(a, b, acc);
        }
        store_c_rm(tmpl, HW, mr, wc, lane, acc);
    }
    __syncthreads();

    // GEMM2: y[h][w] = sum_m B1[h][m] * tmp[m][w].  Tiles: 8 x 8 = 64.
    float* ydst = y + (size_t)bc * HW * HW;
    for (int t = wave; t < 64; t += 8) {
        const int hr = (t >> 3) << 4;
        const int wc = (t & 7) << 4;
        v8f acc = {};
        for (int kb = 0; kb < MM; kb += 4) {
            v2f a = load_a_rm(B1, MM, hr, kb, lane);
            v2f b = load_b_rm(tmpl, HW, kb, wc, lane);       // ds_load
            acc = wmma4(a, b, acc);
        }
        store_c_rm(ydst, HW, hr, wc, lane, acc);
    }
}

extern "C" void kernel_launch(void* const* d_in, const int* in_sizes, int n_in,
                              void* d_out, int out_size, void* d_ws, size_t ws_size,
                              hipStream_t stream) {
    const float* u  = (const float*)d_in[0];   // [16,32,128,128]
    const float* A1 = (const float*)d_in[1];   // [128,128]
    const float* A2 = (const float*)d_in[2];   // [128,128]
    const float* B1 = (const float*)d_in[3];   // [128,64]
    const float* B2 = (const float*)d_in[4];   // [128,64]
    const float* W  = (const float*)d_in[5];   // [288,32,4096]
    float* y = (float*)d_out;                  // [16,32,128,128]

    float* bp   = (float*)d_ws;                            // [512,64,64]  (8 MB)
    float* outm = bp + (size_t)BATCH * CIN * MM * MM;      // [16,32,4096] (8 MB)

    const size_t lds1 = (size_t)(HW * HW + HW * MM) * sizeof(float);  // 96 KB
    const size_t lds3 = (size_t)(MM * MM + MM * HW) * sizeof(float);  // 48 KB

    k1_spectral_fwd<<<dim3(BATCH * CIN),  dim3(256), lds1, stream>>>(u, A1, A2, bp);
    k2_pointwise   <<<dim3(X_N / 8),      dim3(256), 0,    stream>>>(bp, W, outm);
    k3_spectral_bwd<<<dim3(BATCH * COUT), dim3(256), lds3, stream>>>(outm, B1, B2, y);
}